// CompositeEmbeddingA_18219251269948
// MI455X (gfx1250) — compile-verified
//
#include <hip/hip_runtime.h>

typedef __attribute__((ext_vector_type(2))) float v2f;
typedef __attribute__((ext_vector_type(8))) float v8f;

#define BB      8
#define SS      37448
#define EE      256
#define OUTROWS 5704

// ---------------------------------------------------------------------------
// Weight transpose with K-pair interleave:
//   wtp[((kc>>1)*256 + o)*2 + (kc&1)] = w[(o*256 + i)*s + k],  kc = k*256 + i
// A lane's B-fragment (rows kc, kc+1 at one column) becomes a single
// 8-byte load; lanes 0-15 read a contiguous 128B segment (coalesced).
// ---------------------------------------------------------------------------
__global__ void transpose_w_kernel(const float* __restrict__ w,
                                   float* __restrict__ wtp, int s) {
  int total = s * EE * EE;
  int idx = blockIdx.x * blockDim.x + threadIdx.x;
  if (idx >= total) return;
  int r  = idx & 1;                // low bit of kc
  int o  = (idx >> 1) & 255;       // output column
  int kp = idx >> 9;               // kc >> 1
  int i  = ((kp & 127) << 1) | r;  // embedding dim
  int k  = kp >> 7;                // conv tap
  wtp[idx] = w[(o * EE + i) * s + k];
}

// ---------------------------------------------------------------------------
// Layers 0..2 (no conv): out[b, j, e] = sum of 5 embedding gathers.
// One block per (b, token j in 0..583), one thread per e.
// ---------------------------------------------------------------------------
__global__ void __launch_bounds__(256) embed_small_kernel(
    const int* __restrict__ value, const int* __restrict__ depth,
    const int* __restrict__ position, const float* __restrict__ val_emb,
    const float* __restrict__ dep_emb, const float* __restrict__ pos_emb,
    float* __restrict__ out) {
  int b = blockIdx.x / 584;
  int j = blockIdx.x % 584;                 // token index == output row (layers 0-2)
  int li = (j < 8) ? 0 : ((j < 72) ? 1 : 2);
  int e = threadIdx.x;

  long t = (long)b * SS + j;
  int v  = value[t];
  int d  = depth[t];
  const int* pp = position + t * 3;
  int p0 = pp[0], p1 = pp[1], p2 = pp[2];

  const float* ve = val_emb + (size_t)li * 17 * EE;
  const float* de = dep_emb + (size_t)li * 33 * EE;
  const float* pe = pos_emb + (size_t)li * 3 * 64 * EE;

  float x = ve[v * EE + e] + de[d * EE + e] +
            pe[(0 * 64 + p0) * EE + e] +
            pe[(1 * 64 + p1) * EE + e] +
            pe[(2 * 64 + p2) * EE + e];
  out[((size_t)b * OUTROWS + j) * EE + e] = x;
}

// ---------------------------------------------------------------------------
// Fused embedding + blockwise-conv GEMM using V_WMMA_F32_16X16X4_F32.
// Block: 256 threads (8 wave32s). M-tile = 32 output rows (two 16-row
// A sub-tiles), N = 256 (each wave owns 2 N-tiles of 16 columns).
// Per K-step: 2 LDS b64 (A frags) + 2 global b64 (B frags) feed 4 WMMA
// -> 1 memory instruction per matrix instruction.
// ---------------------------------------------------------------------------
__global__ void __launch_bounds__(256) embed_conv_wmma_kernel(
    const int* __restrict__ value, const int* __restrict__ depth,
    const int* __restrict__ position, const float* __restrict__ val_emb,
    const float* __restrict__ dep_emb, const float* __restrict__ pos_emb,
    const float* __restrict__ wtp,    // K-pair-interleaved weights
    const float* __restrict__ bias,   // [256]
    float* __restrict__ out,
    int li, int s, int L_out, int tok_off, int out_row_off) {
  __shared__ float As[32][260];       // stride 260: conflict-free b64 column reads

  const int tid  = threadIdx.x;
  const int wave = tid >> 5;          // 0..7
  const int lane = tid & 31;

  const float* ve  = val_emb + (size_t)li * 17 * EE;
  const float* de  = dep_emb + (size_t)li * 33 * EE;
  const float* pe0 = pos_emb + (size_t)li * 3 * 64 * EE;
  const float* pe1 = pe0 + (size_t)64 * EE;
  const float* pe2 = pe1 + (size_t)64 * EE;

  v8f acc00 = {};                     // (A rows 0-15 , cols wave*32 + 0..15)
  v8f acc01 = {};                     // (A rows 0-15 , cols wave*32 + 16..31)
  v8f acc10 = {};                     // (A rows 16-31, cols wave*32 + 0..15)
  v8f acc11 = {};                     // (A rows 16-31, cols wave*32 + 16..31)

  const int m  = lane & 15;           // A-frag row within sub-tile / B-frag column
  const int kh = lane >> 4;           // K half-select for A/B fragments

  for (int ks = 0; ks < s; ++ks) {
    // ---- stage 32 embedding rows into LDS (each thread: 32 elements) ----
    {
      int rl = tid >> 3;              // row-in-tile 0..31
      int e0 = (tid & 7) << 5;        // 32-element chunk
      int row = blockIdx.x * 32 + rl; // global output row
      int batch = row / L_out;
      int l = row - batch * L_out;
      long t = (long)batch * SS + tok_off + l * s + ks;
      int v = value[t];
      int d = depth[t];
      const int* pp = position + t * 3;
      int p0 = pp[0], p1 = pp[1], p2 = pp[2];
      const float* vrow  = ve  + (size_t)v  * EE;
      const float* drow  = de  + (size_t)d  * EE;
      const float* prow0 = pe0 + (size_t)p0 * EE;
      const float* prow1 = pe1 + (size_t)p1 * EE;
      const float* prow2 = pe2 + (size_t)p2 * EE;
#pragma unroll
      for (int q = 0; q < 32; ++q) {
        int e = e0 + q;
        As[rl][e] = vrow[e] + drow[e] + prow0[e] + prow1[e] + prow2[e];
      }
    }
    __syncthreads();

    // ---- WMMA sweep over K' = 256 (64 steps of K=4) ----
    const int col0 = (wave << 5) + m; // column for N-tile 0 B-frag
    const float* wbase = wtp + ((size_t)(ks * 128) * 256) * 2;
#pragma unroll 4
    for (int kk = 0; kk < 64; ++kk) {
      int ca = (kk << 2) + (kh << 1);
      // A fragments: lanes 0-15 carry K=0,1 ; lanes 16-31 K=2,3 (ISA §7.12.2)
      v2f a0 = *(const v2f*)&As[m][ca];
      v2f a1 = *(const v2f*)&As[16 + m][ca];
      // B fragments: rows (ks*256 + 4kk + 2kh) + {0,1}, one b64 per fragment
      const float* wrow = wbase + (size_t)((kk << 1) + kh) * 512;
      v2f b0 = *(const v2f*)&wrow[col0 << 1];
      v2f b1 = *(const v2f*)&wrow[(col0 + 16) << 1];
      acc00 = __builtin_amdgcn_wmma_f32_16x16x4_f32(
          false, a0, false, b0, (short)0, acc00, false, false);
      acc01 = __builtin_amdgcn_wmma_f32_16x16x4_f32(
          false, a0, false, b1, (short)0, acc01, false, false);
      acc10 = __builtin_amdgcn_wmma_f32_16x16x4_f32(
          false, a1, false, b0, (short)0, acc10, false, false);
      acc11 = __builtin_amdgcn_wmma_f32_16x16x4_f32(
          false, a1, false, b1, (short)0, acc11, false, false);
    }
    __syncthreads();
  }

  // ---- epilogue: bias + store (C/D layout: m = v + 8*(lane>=16), n = lane&15)
  const int n  = lane & 15;
  const int g  = lane >> 4;
  const int cA = (wave << 5) + n;
  const int cB = cA + 16;
  const float biasA = bias[cA];
  const float biasB = bias[cB];
#pragma unroll
  for (int v = 0; v < 8; ++v) {
    int mrow = v + (g << 3);
    // sub-tile 0 (A rows 0-15)
    {
      int row = blockIdx.x * 32 + mrow;
      int batch = row / L_out;
      int l = row - batch * L_out;
      size_t o = ((size_t)batch * OUTROWS + out_row_off + l) * EE;
      out[o + cA] = acc00[v] + biasA;
      out[o + cB] = acc01[v] + biasB;
    }
    // sub-tile 1 (A rows 16-31)
    {
      int row = blockIdx.x * 32 + 16 + mrow;
      int batch = row / L_out;
      int l = row - batch * L_out;
      size_t o = ((size_t)batch * OUTROWS + out_row_off + l) * EE;
      out[o + cA] = acc10[v] + biasA;
      out[o + cB] = acc11[v] + biasB;
    }
  }
}

// ---------------------------------------------------------------------------
extern "C" void kernel_launch(void* const* d_in, const int* in_sizes, int n_in,
                              void* d_out, int out_size, void* d_ws, size_t ws_size,
                              hipStream_t stream) {
  (void)in_sizes; (void)n_in; (void)out_size; (void)ws_size;
  const int*   value    = (const int*)d_in[0];
  const int*   depth    = (const int*)d_in[1];
  const int*   position = (const int*)d_in[2];
  const float* val_emb  = (const float*)d_in[3];
  const float* dep_emb  = (const float*)d_in[4];
  const float* pos_emb  = (const float*)d_in[5];
  const float* conv4_w  = (const float*)d_in[6];
  const float* conv4_b  = (const float*)d_in[7];
  const float* conv8_w  = (const float*)d_in[8];
  const float* conv8_b  = (const float*)d_in[9];
  float* out = (float*)d_out;

  // Workspace: K-pair-interleaved transposed conv weights. Needs 3 MB.
  float* wtp4 = (float*)d_ws;                   // 1024*256 f32 = 1 MB
  float* wtp8 = wtp4 + 1024 * 256;              // 2048*256 f32 = 2 MB

  transpose_w_kernel<<<(4 * EE * EE + 255) / 256, 256, 0, stream>>>(conv4_w, wtp4, 4);
  transpose_w_kernel<<<(8 * EE * EE + 255) / 256, 256, 0, stream>>>(conv8_w, wtp8, 8);

  // Layers 0-2: embedding only (8+64+512 = 584 rows per sample)
  embed_small_kernel<<<BB * 584, 256, 0, stream>>>(
      value, depth, position, val_emb, dep_emb, pos_emb, out);

  // Layer 3: s=4, L_out=1024, token offset 584, out rows 584..1607  (M=8192 -> 256 blocks)
  embed_conv_wmma_kernel<<<256, 256, 0, stream>>>(
      value, depth, position, val_emb, dep_emb, pos_emb,
      wtp4, conv4_b, out, 3, 4, 1024, 584, 584);

  // Layer 4: s=8, L_out=4096, token offset 4680, out rows 1608..5703 (M=32768 -> 1024 blocks)
  embed_conv_wmma_kernel<<<1024, 256, 0, stream>>>(
      value, depth, position, val_emb, dep_emb, pos_emb,
      wtp8, conv8_b, out, 4, 8, 4096, 4680, 1608);
}